// NADE_1855425872245
// MI455X (gfx1250) — compile-verified
//
#include <hip/hip_runtime.h>
#include <stdint.h>

// NADE forward: out[b,k] = V[k,:] . sigmoid( sum_{j<k} W[:,j]*x[b,j] + bias ), out[:,783]=0
// B=256, H=256, D=784. fp32 throughout (783-step cumsum => must stay f32).
// One workgroup per 16-batch tile; 16 waves (wave32) each own a 16-row H slice held as a
// V_WMMA_F32_16X16X4_F32 accumulator tile. Sequential per-pixel scan runs on VALU; the
// chunk-level carry advance runs on WMMA; W/x chunk staging is double-buffered with
// async global->LDS copies (ASYNCcnt) so it overlaps the scan of the previous chunk.

#define B_   256
#define H_   256
#define D_   784
#define DM1  783
#define BT   16     // batch tile (WMMA N)
#define NW   16     // waves per block (H_/16)
#define CK   16     // k-chunk length

typedef __attribute__((ext_vector_type(2))) float v2f;
typedef __attribute__((ext_vector_type(8))) float v8f;

// ---- CDNA5 async global->LDS copy (portable inline asm; ASYNCcnt-tracked) ----
__device__ __forceinline__ void async_g2l_b32(float* lds_dst, const float* gsrc) {
    // VDST = 32-bit LDS byte address (low 32 bits of the flat LDS-aperture pointer),
    // VADDR = 64-bit global address, SADDR = off.  ISA 15.18.3 opcode 96.
    uint32_t loff = (uint32_t)(uintptr_t)lds_dst;
    asm volatile("global_load_async_to_lds_b32 %0, %1, off"
                 :: "v"(loff), "v"(gsrc) : "memory");
}
__device__ __forceinline__ void wait_async0() {
    asm volatile("s_wait_asynccnt 0" ::: "memory");
}

__global__ __launch_bounds__(NW * 32)
void nade_scan_kernel(const float* __restrict__ x,   // (B, D-1)
                      const float* __restrict__ W,   // (H, D)
                      const float* __restrict__ bias,// (H,)
                      const float* __restrict__ V,   // (D, H)
                      float* __restrict__ out)       // (B, D)
{
    __shared__ float Wt[2][CK * H_];     // Wt[buf][j][h] = W[h, kc+j]  (2 x 16 KB)
    __shared__ float xs[2][BT * 17];     // xs[buf][b][j] (pad 17 vs bank conflicts)
    __shared__ float red[NW * CK * BT];  // per-wave partial logits (16 KB)

    const int t    = threadIdx.x;
    const int lane = t & 31;
    const int w    = t >> 5;           // wave id -> H slice
    const int half = lane >> 4;        // 0: tile rows 0..7, 1: rows 8..15 (f32 C layout)
    const int bcol = lane & 15;        // batch column / A-matrix row index
    const int h0   = w * 16;
    const int b0   = blockIdx.x * BT;

    // bias for this lane's 8 state rows (C/D layout: vgpr i -> row half*8+i)
    float bs[8];
    {
        const float* bp = bias + h0 + half * 8;
        float4 t0 = *(const float4*)bp;
        float4 t1 = *(const float4*)(bp + 4);
        bs[0]=t0.x; bs[1]=t0.y; bs[2]=t0.z; bs[3]=t0.w;
        bs[4]=t1.x; bs[5]=t1.y; bs[6]=t1.z; bs[7]=t1.w;
    }

    // output column D-1 is defined to be zero (harness poisons d_out)
    if (t < BT) out[(size_t)(b0 + t) * D_ + (D_ - 1)] = 0.0f;

    v8f S = {0.f,0.f,0.f,0.f,0.f,0.f,0.f,0.f};  // carried exclusive-prefix state tile

    const int NC = (DM1 + CK - 1) / CK;  // 49 chunks (last one covers j=768..782)

    // ---- async stage of one chunk into LDS buffer `buf` (issue only, no wait) ----
    auto stage = [&](int c, int buf) {
        const int kc = c * CK;
        #pragma unroll
        for (int r = 0; r < (CK * H_) / (NW * 32); ++r) {   // 8 x b32 per thread
            int idx = r * (NW * 32) + t;
            int j = idx >> 8, h = idx & (H_ - 1);
            async_g2l_b32(&Wt[buf][j * H_ + h], &W[(size_t)h * D_ + kc + j]);
        }
        if (t < BT * CK) {
            int b = t >> 4, j = t & 15;
            int jg = kc + j;
            if (jg < DM1)
                async_g2l_b32(&xs[buf][b * 17 + j], &x[(size_t)(b0 + b) * DM1 + jg]);
            else
                xs[buf][b * 17 + j] = 0.0f;   // tail pixel: plain ds_store of zero
        }
    };

    stage(0, 0);   // prologue: chunk 0 in flight

    for (int c = 0; c < NC; ++c) {
        const int kc  = c * CK;
        const int cur = c & 1;

        wait_async0();      // my async writes for chunk c done
        __syncthreads();    // all waves' writes visible; also protects red + alt buffer

        if (c + 1 < NC) stage(c + 1, cur ^ 1);   // overlaps with the scan below

        // ---- VALU critical path: sequential exclusive-prefix scan of this chunk ----
        float s[8];
        #pragma unroll
        for (int i = 0; i < 8; ++i) s[i] = S[i];

        for (int k = 0; k < CK; ++k) {
            const int j = kc + k;
            if (j >= DM1) break;                       // uniform: trims only last chunk
            if (k > 0) {                               // add term j-1 (exclusive prefix)
                const float* wp = &Wt[cur][(k - 1) * H_ + h0 + half * 8];
                float4 w0 = *(const float4*)wp;        // broadcast b128 LDS reads
                float4 w1 = *(const float4*)(wp + 4);
                float xv = xs[cur][bcol * 17 + (k - 1)];
                s[0] = fmaf(w0.x, xv, s[0]);  s[1] = fmaf(w0.y, xv, s[1]);
                s[2] = fmaf(w0.z, xv, s[2]);  s[3] = fmaf(w0.w, xv, s[3]);
                s[4] = fmaf(w1.x, xv, s[4]);  s[5] = fmaf(w1.y, xv, s[5]);
                s[6] = fmaf(w1.z, xv, s[6]);  s[7] = fmaf(w1.w, xv, s[7]);
            }
            // V row k is contiguous in memory (V is k-major) -> L2-resident broadcast
            const float* vp = V + (size_t)j * H_ + h0 + half * 8;
            float4 v0 = *(const float4*)vp;
            float4 v1 = *(const float4*)(vp + 4);
            const float vv[8] = {v0.x,v0.y,v0.z,v0.w,v1.x,v1.y,v1.z,v1.w};
            float p = 0.f;
            #pragma unroll
            for (int i = 0; i < 8; ++i) {
                float z = s[i] + bs[i];
                float a = 1.0f / (1.0f + __expf(-z));  // v_exp_f32 (TRANS)
                p = fmaf(vv[i], a, p);
            }
            p += __shfl_xor(p, 16, 32);                // combine the two half-wave row groups
            if (lane < 16) red[(w * CK + k) * BT + lane] = p;
        }

        // ---- WMMA path: advance carried state, S += W_chunk(16x16) @ Xc(16x16) ----
        // 4 steps of V_WMMA_F32_16X16X4_F32; co-executes with surrounding VALU (XDL).
        if (c + 1 < NC) {
            #pragma unroll
            for (int step = 0; step < 4; ++step) {
                const int j0   = kc + step * 4;
                const int koff = half * 2;             // lanes 0-15: K{0,1}; 16-31: K{2,3}
                v2f a, bb;
                // A (16x4): row = bcol (=h0+bcol), two consecutive K values -> b64 load
                const float* ap = W + (size_t)(h0 + bcol) * D_ + j0 + koff;
                a[0] = ap[0];
                a[1] = ap[1];
                // B (4x16): column = bcol, same K striping
                bb[0] = xs[cur][bcol * 17 + step * 4 + koff];
                bb[1] = xs[cur][bcol * 17 + step * 4 + koff + 1];
                S = __builtin_amdgcn_wmma_f32_16x16x4_f32(
                        /*neg_a=*/false, a, /*neg_b=*/false, bb,
                        /*c_mod=*/(short)0, S, /*reuse_a=*/false, /*reuse_b=*/false);
            }
        }

        __syncthreads();

        // ---- deterministic cross-wave reduction (fixed order, no float atomics) ----
        if (t < CK * BT) {
            int k = t >> 4, bb2 = t & 15;
            int j = kc + k;
            if (j < DM1) {
                float sum = 0.f;
                #pragma unroll
                for (int w2 = 0; w2 < NW; ++w2)
                    sum += red[(w2 * CK + k) * BT + bb2];
                out[(size_t)(b0 + bb2) * D_ + j] = sum;
            }
        }
    }
}

extern "C" void kernel_launch(void* const* d_in, const int* in_sizes, int n_in,
                              void* d_out, int out_size, void* d_ws, size_t ws_size,
                              hipStream_t stream) {
    const float* x  = (const float*)d_in[0];  // (256, 783)
    const float* W  = (const float*)d_in[1];  // (256, 784)
    const float* bv = (const float*)d_in[2];  // (256,)
    const float* V  = (const float*)d_in[3];  // (784, 256)
    float* out = (float*)d_out;               // (256, 784)
    nade_scan_kernel<<<dim3(B_ / BT), dim3(NW * 32), 0, stream>>>(x, W, bv, V, out);
    (void)in_sizes; (void)n_in; (void)out_size; (void)d_ws; (void)ws_size;
}